// SignNetLayer_19138374271543
// MI455X (gfx1250) — compile-verified
//
#include <hip/hip_runtime.h>
#include <hip/hip_bf16.h>

#define KEIG 4
#define HID  64

typedef float v2f __attribute__((ext_vector_type(2)));
typedef float v8f __attribute__((ext_vector_type(8)));

// fp32 WMMA: D(16x16) = A(16x4) * B(4x16) + C    (V_WMMA_F32_16X16X4_F32)
__device__ __forceinline__ v8f wmma4(v2f a, v2f b, v8f c) {
  return __builtin_amdgcn_wmma_f32_16x16x4_f32(false, a, false, b, (short)0, c,
                                               false, false);
}

// ---------------------------------------------------------------------------
// Layer 0 edge pass: F_in = 1.  e_emb = edge_attr@We0 + be0 (scalar),
// msg = relu(+-x[src] + e_emb), scatter-add into agg (both signs, all K).
// ---------------------------------------------------------------------------
__global__ void k_edge_l0(const float* __restrict__ x,
                          const float* __restrict__ edge_attr,
                          const int* __restrict__ src, const int* __restrict__ dst,
                          const float* __restrict__ We, const float* __restrict__ be,
                          float* __restrict__ ap, float* __restrict__ an,
                          int E, int N) {
  int e = blockIdx.x * blockDim.x + threadIdx.x;
  if (e >= E) return;
  float emb = be[0];
#pragma unroll
  for (int j = 0; j < 32; ++j) emb += edge_attr[(size_t)e * 32 + j] * We[j];
  int s = src[e], d = dst[e];
#pragma unroll
  for (int k = 0; k < KEIG; ++k) {
    float hv = x[(size_t)k * N + s];
    atomicAdd(&ap[(size_t)k * N + d], fmaxf( hv + emb, 0.f));
    atomicAdd(&an[(size_t)k * N + d], fmaxf(-hv + emb, 0.f));
  }
}

// ---------------------------------------------------------------------------
// Layers 1/2 edge pass (F = 64).  Each wave owns a 16-edge tile:
//   e_emb[16x64] = edge_attr[16x32] @ We[32x64] + be   via 32 chained WMMAs,
// staged in LDS, then fused gather/relu/scatter for both signs x 4 channels.
// block = 128 threads = 4 waves.
// ---------------------------------------------------------------------------
__global__ void k_edge_msg(const float* __restrict__ hp, const float* __restrict__ hn,
                           float* __restrict__ ap, float* __restrict__ an,
                           const float* __restrict__ edge_attr,
                           const int* __restrict__ src, const int* __restrict__ dst,
                           const float* __restrict__ We, const float* __restrict__ be,
                           int E, int N) {
  __shared__ float sWe[32 * HID];       // 8 KB
  __shared__ float sE[4][16 * HID];     // 16 KB e_emb tiles (one per wave)

  const int tid  = threadIdx.x;
  const int lane = tid & 31;
  const int wave = tid >> 5;
  for (int i = tid; i < 32 * HID; i += 128) sWe[i] = We[i];
  __syncthreads();

  const int e0  = (blockIdx.x * 4 + wave) * 16;
  const int mlo = lane & 15;
  const int hi  = lane >> 4;

  // A fragments: 16 edges x 32 edge-features
  const int e  = e0 + mlo;
  const bool ev = e < E;
  float ea[16];
#pragma unroll
  for (int kb = 0; kb < 8; ++kb)
#pragma unroll
    for (int j = 0; j < 2; ++j) {
      int k = 4 * kb + j + 2 * hi;
      ea[2 * kb + j] = ev ? edge_attr[(size_t)e * 32 + k] : 0.f;
    }

#pragma unroll
  for (int nt = 0; nt < 4; ++nt) {
    v8f acc;
    float bias = be[nt * 16 + mlo];
#pragma unroll
    for (int i = 0; i < 8; ++i) acc[i] = bias;
#pragma unroll
    for (int kb = 0; kb < 8; ++kb) {
      v2f a; a[0] = ea[2 * kb]; a[1] = ea[2 * kb + 1];
      v2f b;
      b[0] = sWe[(4 * kb + 0 + 2 * hi) * HID + nt * 16 + mlo];
      b[1] = sWe[(4 * kb + 1 + 2 * hi) * HID + nt * 16 + mlo];
      acc = wmma4(a, b, acc);
    }
#pragma unroll
    for (int i = 0; i < 8; ++i)
      sE[wave][(i + 8 * hi) * HID + nt * 16 + mlo] = acc[i];
  }
  __syncthreads();

  // scatter phase: e_emb shared across both signs and all K channels
  for (int t = 0; t < 16; ++t) {
    int ee = e0 + t;
    if (ee >= E) break;
    int se = src[ee], de = dst[ee];
#pragma unroll
    for (int k = 0; k < KEIG; ++k) {
      const float* hps = hp + ((size_t)k * N + se) * HID;
      const float* hns = hn + ((size_t)k * N + se) * HID;
      float* apd = ap + ((size_t)k * N + de) * HID;
      float* and_ = an + ((size_t)k * N + de) * HID;
#pragma unroll
      for (int f = lane; f < HID; f += 32) {
        float eef = sE[wave][t * HID + f];
        atomicAdd(&apd[f],  fmaxf(hps[f] + eef, 0.f));
        atomicAdd(&and_[f], fmaxf(hns[f] + eef, 0.f));
      }
    }
  }
}

// ---------------------------------------------------------------------------
// Fused per-row MLP:  out = [relu?]( relu( (scale*h + agg) @ W1 + b1 ) @ W2 + b2 )
// block = 128 threads = 4 waves; each wave computes a 16-row tile.
// Both GEMMs use f32 WMMA (K reduced in steps of 4).
// ---------------------------------------------------------------------------
template <int F_IN, int F_OUT, bool USE_AGG, bool RELU_OUT>
__global__ void k_node_mlp(const float* __restrict__ h, const float* __restrict__ agg,
                           float* __restrict__ out, float h_scale,
                           const float* __restrict__ W1, const float* __restrict__ b1,
                           const float* __restrict__ W2, const float* __restrict__ b2,
                           int rows) {
  __shared__ float sW1[F_IN * HID];
  __shared__ float sW2[HID * F_OUT];
  __shared__ float sH[4][16 * HID];

  const int tid  = threadIdx.x;
  const int lane = tid & 31;
  const int wave = tid >> 5;
  for (int i = tid; i < F_IN * HID; i += 128) sW1[i] = W1[i];
  for (int i = tid; i < HID * F_OUT; i += 128) sW2[i] = W2[i];
  __syncthreads();

  const int row0 = (blockIdx.x * 4 + wave) * 16;
  const int mlo  = lane & 15;
  const int hi   = lane >> 4;

  if constexpr (F_IN == 1) {
    // hidden = relu(z * W1[n] + b1[n])  (outer product, no GEMM needed)
    for (int idx = lane; idx < 16 * HID; idx += 32) {
      int mm = idx >> 6, nn = idx & 63;
      int rr = row0 + mm;
      float z = 0.f;
      if (rr < rows) {
        z = h_scale * h[rr];
        if (USE_AGG) z += agg[rr];
      }
      sH[wave][idx] = fmaxf(z * sW1[nn] + b1[nn], 0.f);
    }
  } else {
    constexpr int NS = F_IN / 4;
    float za[2 * NS];
    const int rr  = row0 + mlo;
    const bool rv = rr < rows;
#pragma unroll
    for (int kb = 0; kb < NS; ++kb)
#pragma unroll
      for (int j = 0; j < 2; ++j) {
        int k = 4 * kb + j + 2 * hi;
        float z = 0.f;
        if (rv) {
          z = h_scale * h[(size_t)rr * F_IN + k];
          if (USE_AGG) z += agg[(size_t)rr * F_IN + k];
        }
        za[2 * kb + j] = z;
      }
#pragma unroll
    for (int nt = 0; nt < 4; ++nt) {
      v8f acc;
      float bias = b1[nt * 16 + mlo];
#pragma unroll
      for (int i = 0; i < 8; ++i) acc[i] = bias;
#pragma unroll
      for (int kb = 0; kb < NS; ++kb) {
        v2f a; a[0] = za[2 * kb]; a[1] = za[2 * kb + 1];
        v2f b;
        b[0] = sW1[(4 * kb + 0 + 2 * hi) * HID + nt * 16 + mlo];
        b[1] = sW1[(4 * kb + 1 + 2 * hi) * HID + nt * 16 + mlo];
        acc = wmma4(a, b, acc);
      }
#pragma unroll
      for (int i = 0; i < 8; ++i)
        sH[wave][(i + 8 * hi) * HID + nt * 16 + mlo] = fmaxf(acc[i], 0.f);
    }
  }
  __syncthreads();

  // GEMM2: H[16x64] @ W2[64xF_OUT] + b2
  constexpr int NT2 = F_OUT / 16;
#pragma unroll
  for (int nt = 0; nt < NT2; ++nt) {
    v8f acc;
    float bias = b2[nt * 16 + mlo];
#pragma unroll
    for (int i = 0; i < 8; ++i) acc[i] = bias;
#pragma unroll
    for (int kb = 0; kb < HID / 4; ++kb) {
      v2f a;
      a[0] = sH[wave][mlo * HID + 4 * kb + 0 + 2 * hi];
      a[1] = sH[wave][mlo * HID + 4 * kb + 1 + 2 * hi];
      v2f b;
      b[0] = sW2[(4 * kb + 0 + 2 * hi) * F_OUT + nt * 16 + mlo];
      b[1] = sW2[(4 * kb + 1 + 2 * hi) * F_OUT + nt * 16 + mlo];
      acc = wmma4(a, b, acc);
    }
#pragma unroll
    for (int i = 0; i < 8; ++i) {
      int rr = row0 + i + 8 * hi;
      if (rr < rows) {
        float v = acc[i];
        if (RELU_OUT) v = fmaxf(v, 0.f);
        out[(size_t)rr * F_OUT + nt * 16 + mlo] = v;
      }
    }
  }
}

// ---------------------------------------------------------------------------
// pooled[n, f] = sum_k ( phi_pos[k,n,f] + phi_neg[k,n,f] ),  f < 32
// ---------------------------------------------------------------------------
__global__ void k_pool(const float* __restrict__ hp, const float* __restrict__ hn,
                       float* __restrict__ pooled, int N) {
  int i = blockIdx.x * blockDim.x + threadIdx.x;
  if (i >= N * 32) return;
  int n = i >> 5, f = i & 31;
  float s = 0.f;
#pragma unroll
  for (int k = 0; k < KEIG; ++k)
    s += hp[((size_t)k * N + n) * 32 + f] + hn[((size_t)k * N + n) * 32 + f];
  pooled[(size_t)n * 32 + f] = s;
}

// ---------------------------------------------------------------------------
extern "C" void kernel_launch(void* const* d_in, const int* in_sizes, int n_in,
                              void* d_out, int out_size, void* d_ws, size_t ws_size,
                              hipStream_t stream) {
  (void)n_in; (void)out_size; (void)ws_size;
  const float* x         = (const float*)d_in[0];      // [K, N, 1]
  const float* edge_attr = (const float*)d_in[1];      // [E, 32]
  const int*   ei        = (const int*)d_in[2];        // [2, E]
  const float* l0_We = (const float*)d_in[3],  *l0_be = (const float*)d_in[4];
  const float* l0_W1 = (const float*)d_in[5],  *l0_b1 = (const float*)d_in[6];
  const float* l0_W2 = (const float*)d_in[7],  *l0_b2 = (const float*)d_in[8];
  const float* l1_We = (const float*)d_in[9],  *l1_be = (const float*)d_in[10];
  const float* l1_W1 = (const float*)d_in[11], *l1_b1 = (const float*)d_in[12];
  const float* l1_W2 = (const float*)d_in[13], *l1_b2 = (const float*)d_in[14];
  const float* l2_We = (const float*)d_in[15], *l2_be = (const float*)d_in[16];
  const float* l2_W1 = (const float*)d_in[17], *l2_b1 = (const float*)d_in[18];
  const float* l2_W2 = (const float*)d_in[19], *l2_b2 = (const float*)d_in[20];
  const float* Wr1 = (const float*)d_in[21], *br1 = (const float*)d_in[22];
  const float* Wr2 = (const float*)d_in[23], *br2 = (const float*)d_in[24];
  float* outp = (float*)d_out;

  const int N = in_sizes[0] / KEIG;       // 20000
  const int E = in_sizes[1] / 32;         // 640000
  const size_t KN   = (size_t)KEIG * N;   // 80000 rows
  const size_t KN64 = KN * HID;

  float* ws = (float*)d_ws;
  float* ap  = ws;                // agg  (pos)
  float* an  = ap  + KN64;        // agg  (neg)
  float* hp  = an  + KN64;        // node features (pos)
  float* hn  = hp  + KN64;        // node features (neg)
  float* hp2 = hn  + KN64;
  float* hn2 = hp2 + KN64;
  float* pooled = hn2 + KN64;     // [N, 32]

  const int rowsKN = (int)KN;
  dim3 blk128(128);
  const int mlp_grid  = (rowsKN + 63) / 64;   // 4 waves x 16 rows per block
  const int rho_grid  = (N + 63) / 64;
  const int etiles    = (E + 15) / 16;
  const int emsg_grid = (etiles + 3) / 4;

  // ---- layer 0 (F_in = 1) ----
  hipMemsetAsync(ap, 0, KN * sizeof(float), stream);
  hipMemsetAsync(an, 0, KN * sizeof(float), stream);
  k_edge_l0<<<(E + 255) / 256, 256, 0, stream>>>(x, edge_attr, ei, ei + E,
                                                 l0_We, l0_be, ap, an, E, N);
  k_node_mlp<1, 64, true, true><<<mlp_grid, blk128, 0, stream>>>(
      x, ap, hp,  1.0f, l0_W1, l0_b1, l0_W2, l0_b2, rowsKN);
  k_node_mlp<1, 64, true, true><<<mlp_grid, blk128, 0, stream>>>(
      x, an, hn, -1.0f, l0_W1, l0_b1, l0_W2, l0_b2, rowsKN);

  // ---- layer 1 (64 -> 64, relu) ----
  hipMemsetAsync(ap, 0, KN64 * sizeof(float), stream);
  hipMemsetAsync(an, 0, KN64 * sizeof(float), stream);
  k_edge_msg<<<emsg_grid, blk128, 0, stream>>>(hp, hn, ap, an, edge_attr,
                                               ei, ei + E, l1_We, l1_be, E, N);
  k_node_mlp<64, 64, true, true><<<mlp_grid, blk128, 0, stream>>>(
      hp, ap, hp2, 1.0f, l1_W1, l1_b1, l1_W2, l1_b2, rowsKN);
  k_node_mlp<64, 64, true, true><<<mlp_grid, blk128, 0, stream>>>(
      hn, an, hn2, 1.0f, l1_W1, l1_b1, l1_W2, l1_b2, rowsKN);

  // ---- layer 2 (64 -> 32, no trailing relu) ----
  hipMemsetAsync(ap, 0, KN64 * sizeof(float), stream);
  hipMemsetAsync(an, 0, KN64 * sizeof(float), stream);
  k_edge_msg<<<emsg_grid, blk128, 0, stream>>>(hp2, hn2, ap, an, edge_attr,
                                               ei, ei + E, l2_We, l2_be, E, N);
  k_node_mlp<64, 32, true, false><<<mlp_grid, blk128, 0, stream>>>(
      hp2, ap, hp, 1.0f, l2_W1, l2_b1, l2_W2, l2_b2, rowsKN);
  k_node_mlp<64, 32, true, false><<<mlp_grid, blk128, 0, stream>>>(
      hn2, an, hn, 1.0f, l2_W1, l2_b1, l2_W2, l2_b2, rowsKN);

  // ---- pool over K + sign, then rho MLP -> d_out ----
  k_pool<<<(N * 32 + 255) / 256, 256, 0, stream>>>(hp, hn, pooled, N);
  k_node_mlp<32, 64, false, false><<<rho_grid, blk128, 0, stream>>>(
      pooled, nullptr, outp, 1.0f, Wr1, br1, Wr2, br2, N);
}